// GCN_60404420051602
// MI455X (gfx1250) — compile-verified
//
#include <hip/hip_runtime.h>
#include <hip/hip_bf16.h>

#define N_NODES 50000
#define N_EDGES 600000
#define D_IN    256
#define D       128
#define ROW_TILES (N_NODES / 16)        // 3125, exact
#define TILES_PER_BLK 8                 // 8 waves * 16 rows = 128 rows / block
#define GEMM_GRID ((ROW_TILES + TILES_PER_BLK - 1) / TILES_PER_BLK)  // 391

// LDS row strides (floats): K + 4  ->  stride % 64 == 4, so the 32 lanes'
// ds_load_b64 B-fragment reads land on 32 disjoint bank pairs (no conflicts).
#define EMB_S (D_IN + 4)                // 260
#define LAY_S (D + 4)                   // 132

typedef __attribute__((ext_vector_type(2))) float v2f;
typedef __attribute__((ext_vector_type(8))) float v8f;
typedef __attribute__((ext_vector_type(4))) unsigned int v4u;
typedef __attribute__((ext_vector_type(8))) int v8i;
typedef __attribute__((ext_vector_type(4))) int v4i;

// ------------------------------------------------------------- TDM staging
// Stage a 128 x K (row-major, element = f32) weight matrix W into LDS with a
// padded row stride of K+4 floats, using the Tensor Data Mover.
// D# per CDNA5 ISA:
//   group0: count=1, lds_addr, global_addr[56:0], type=2
//   group1: data_size=2 (4B), pad_enable, pad_interval (K DWORDs),
//           pad_amount=3 (4 DWORDs), tensor_dim0=K, tensor_dim1=128,
//           tile_dim0=K, tile_dim1=128, tensor_dim0_stride=K
// This toolchain's builtin takes the 6-arg form:
//   (uint32x4 g0, int32x8 g1, int32x4, int32x4, int32x8, i32 cpol)
__device__ __forceinline__ void tdm_stage_weights(const float* W, unsigned K,
                                                  unsigned pad_interval_code,
                                                  unsigned lds_byte_off) {
    unsigned long long ga = (unsigned long long)(uintptr_t)W;
    v4u g0;
    g0[0] = 1u;                                            // count=1
    g0[1] = lds_byte_off;                                  // lds_addr (bytes)
    g0[2] = (unsigned)ga;                                  // global_addr lo
    g0[3] = (unsigned)((ga >> 32) & 0x01FFFFFFu) | (2u << 30);  // hi | type=2

    v8i g1;
    g1[0] = (int)((2u << 16) | (1u << 20) |                // data_size=4B, pad_en
                  (pad_interval_code << 22) | (3u << 25)); // pad 4 DWORDs
    g1[1] = (int)((K & 0xFFFFu) << 16);                    // tensor_dim0 lo16
    g1[2] = (int)(((K >> 16) & 0xFFFFu) | (128u << 16));   // dim0 hi16 | dim1 lo16
    g1[3] = (int)(K << 16);                                // dim1 hi16=0 | tile_dim0
    g1[4] = (int)128u;                                     // tile_dim1 (tile_dim2=0)
    g1[5] = (int)K;                                        // tensor_dim0_stride lo32
    g1[6] = 0;                                             // stride hi | dim1_stride lo
    g1[7] = 0;

    v4i gz4 = {0, 0, 0, 0};
    v8i gz8 = {0, 0, 0, 0, 0, 0, 0, 0};
    __builtin_amdgcn_tensor_load_to_lds(g0, g1, gz4, gz4, gz8, 0);
    __builtin_amdgcn_s_wait_tensorcnt(0);
}

// ---------------------------------------------------------------- utilities

__global__ void zero_f_kernel(float* __restrict__ p, int n) {
    int i = blockIdx.x * blockDim.x + threadIdx.x;
    if (i < n) p[i] = 0.0f;
}

__global__ void degree_kernel(const int* __restrict__ dst, float* __restrict__ deg) {
    int e = blockIdx.x * blockDim.x + threadIdx.x;
    if (e < N_EDGES) {
        __hip_atomic_fetch_add(&deg[dst[e]], 1.0f,
                               __ATOMIC_RELAXED, __HIP_MEMORY_SCOPE_AGENT);
    }
}

__global__ void dinv_kernel(float* __restrict__ deg_dinv) {
    int i = blockIdx.x * blockDim.x + threadIdx.x;
    if (i < N_NODES) deg_dinv[i] = rsqrtf(deg_dinv[i] + 1.0f);
}

// out[n][c] = dinv[n]^2 * xw[n][c] + bias[c]   (self-loop term + bias)
__global__ void agg_init_kernel(const float* __restrict__ xw,
                                const float* __restrict__ dinv,
                                const float* __restrict__ bias,
                                float* __restrict__ out) {
    int i = blockIdx.x * blockDim.x + threadIdx.x;   // exact grid = N_NODES*D
    int n = i >> 7;
    int c = i & (D - 1);
    float di = dinv[n];
    out[i] = di * di * xw[i] + bias[c];
}

// one wave per edge: out[d][:] += dinv[s]*dinv[d] * xw[s][:]
__global__ void edge_scatter_kernel(const int* __restrict__ src,
                                    const int* __restrict__ dst,
                                    const float* __restrict__ dinv,
                                    const float* __restrict__ xw,
                                    float* __restrict__ out) {
    int e = (blockIdx.x * blockDim.x + threadIdx.x) >> 5;
    int lane = threadIdx.x & 31;
    if (e >= N_EDGES) return;
    int s = src[e];
    int d = dst[e];
    float nrm = dinv[s] * dinv[d];
    const float4 v = *(const float4*)(xw + (size_t)s * D + lane * 4);
    float* po = out + (size_t)d * D + lane * 4;
    __hip_atomic_fetch_add(po + 0, v.x * nrm, __ATOMIC_RELAXED, __HIP_MEMORY_SCOPE_AGENT);
    __hip_atomic_fetch_add(po + 1, v.y * nrm, __ATOMIC_RELAXED, __HIP_MEMORY_SCOPE_AGENT);
    __hip_atomic_fetch_add(po + 2, v.z * nrm, __ATOMIC_RELAXED, __HIP_MEMORY_SCOPE_AGENT);
    __hip_atomic_fetch_add(po + 3, v.w * nrm, __ATOMIC_RELAXED, __HIP_MEMORY_SCOPE_AGENT);
}

// ---------------------------------------------------------------- WMMA GEMMs
// fp32 WMMA 16x16x4, one wave per 16-row x 128-col output slab.
// A-frag: lane L holds A[L%16][k0 + 2*(L/16) + {0,1}]        (global float2)
// B-frag: lane L holds B[k0 + 2*(L/16) + {0,1}][L%16]        (LDS float2)
//         B[k][n] = W[n][k]  (x = a @ W^T), W staged in LDS, stride K+4
// D: vgpr v -> row v + 8*(L/16), col L%16

// x0[n][:] = embed_table[tokens[n]] @ W^T + bias   (K = 256)
__global__ void embed_gemm_kernel(const int* __restrict__ tokens,
                                  const float* __restrict__ table,  // [VOCAB][256]
                                  const float* __restrict__ W,      // [128][256]
                                  const float* __restrict__ bias,
                                  float* __restrict__ out) {        // [N][128]
    extern __shared__ float smem[];
    const int lane = threadIdx.x & 31;
    const int tile = blockIdx.x * TILES_PER_BLK + (threadIdx.x >> 5);

    if (threadIdx.x < 32) {   // wave 0 issues the TDM copy (EXEC-independent)
        tdm_stage_weights(W, D_IN, /*pad_interval: 256 DW*/ 7u,
                          (unsigned)__builtin_amdgcn_groupstaticsize());
    }
    __syncthreads();

    if (tile < ROW_TILES) {
        const int m  = lane & 15;
        const int kh = (lane >> 4) << 1;          // 0 or 2
        const float* arow = table + (size_t)tokens[tile * 16 + m] * D_IN + kh;

        v8f acc[8];
#pragma unroll
        for (int nt = 0; nt < 8; ++nt) acc[nt] = (v8f){0,0,0,0,0,0,0,0};

        for (int k0 = 0; k0 < D_IN; k0 += 8) {
            v2f a0 = *(const v2f*)(arow + k0);
            v2f a1 = *(const v2f*)(arow + k0 + 4);
            v2f b0[8], b1[8];
#pragma unroll
            for (int nt = 0; nt < 8; ++nt) {
                const float* bp = smem + (nt * 16 + m) * EMB_S + k0 + kh;
                b0[nt] = *(const v2f*)(bp);
                b1[nt] = *(const v2f*)(bp + 4);
            }
#pragma unroll
            for (int nt = 0; nt < 8; ++nt)
                acc[nt] = __builtin_amdgcn_wmma_f32_16x16x4_f32(
                    false, a0, false, b0[nt], (short)0, acc[nt], false, false);
#pragma unroll
            for (int nt = 0; nt < 8; ++nt)
                acc[nt] = __builtin_amdgcn_wmma_f32_16x16x4_f32(
                    false, a1, false, b1[nt], (short)0, acc[nt], false, false);
        }

        const int g = lane >> 4;
#pragma unroll
        for (int nt = 0; nt < 8; ++nt) {
#pragma unroll
            for (int v = 0; v < 8; ++v) {
                int r = v + 8 * g;
                int c = nt * 16 + m;
                out[(size_t)(tile * 16 + r) * D + c] = acc[nt][v] + bias[c];
            }
        }
    }
}

// out[n][:] = act(x[n][:]) @ W^T    (K = 128)
template <bool RELU>
__global__ void node_gemm_kernel(const float* __restrict__ x,  // [N][128]
                                 const float* __restrict__ W,  // [128][128]
                                 float* __restrict__ out) {    // [N][128]
    extern __shared__ float smem[];
    const int lane = threadIdx.x & 31;
    const int tile = blockIdx.x * TILES_PER_BLK + (threadIdx.x >> 5);

    if (threadIdx.x < 32) {
        tdm_stage_weights(W, D, /*pad_interval: 128 DW*/ 6u,
                          (unsigned)__builtin_amdgcn_groupstaticsize());
    }
    __syncthreads();

    if (tile < ROW_TILES) {
        const int m  = lane & 15;
        const int kh = (lane >> 4) << 1;
        const float* arow = x + (size_t)(tile * 16 + m) * D + kh;

        v8f acc[8];
#pragma unroll
        for (int nt = 0; nt < 8; ++nt) acc[nt] = (v8f){0,0,0,0,0,0,0,0};

        for (int k0 = 0; k0 < D; k0 += 8) {
            v2f a0 = *(const v2f*)(arow + k0);
            v2f a1 = *(const v2f*)(arow + k0 + 4);
            if (RELU) {
                a0.x = fmaxf(a0.x, 0.0f);  a0.y = fmaxf(a0.y, 0.0f);
                a1.x = fmaxf(a1.x, 0.0f);  a1.y = fmaxf(a1.y, 0.0f);
            }
            v2f b0[8], b1[8];
#pragma unroll
            for (int nt = 0; nt < 8; ++nt) {
                const float* bp = smem + (nt * 16 + m) * LAY_S + k0 + kh;
                b0[nt] = *(const v2f*)(bp);
                b1[nt] = *(const v2f*)(bp + 4);
            }
#pragma unroll
            for (int nt = 0; nt < 8; ++nt)
                acc[nt] = __builtin_amdgcn_wmma_f32_16x16x4_f32(
                    false, a0, false, b0[nt], (short)0, acc[nt], false, false);
#pragma unroll
            for (int nt = 0; nt < 8; ++nt)
                acc[nt] = __builtin_amdgcn_wmma_f32_16x16x4_f32(
                    false, a1, false, b1[nt], (short)0, acc[nt], false, false);
        }

        const int g = lane >> 4;
#pragma unroll
        for (int nt = 0; nt < 8; ++nt) {
#pragma unroll
            for (int v = 0; v < 8; ++v) {
                int r = v + 8 * g;
                int c = nt * 16 + m;
                out[(size_t)(tile * 16 + r) * D + c] = acc[nt][v];
            }
        }
    }
}

// ---------------------------------------------------------------- launcher

extern "C" void kernel_launch(void* const* d_in, const int* in_sizes, int n_in,
                              void* d_out, int out_size, void* d_ws, size_t ws_size,
                              hipStream_t stream) {
    const int*   node_tokens = (const int*)  d_in[0];
    const int*   edge_index  = (const int*)  d_in[1];   // [2][600000]
    const float* embed_table = (const float*)d_in[2];
    const float* W_node_w    = (const float*)d_in[3];
    const float* W_node_b    = (const float*)d_in[4];
    const float* w1          = (const float*)d_in[5];
    const float* b1          = (const float*)d_in[6];
    const float* w2          = (const float*)d_in[7];
    const float* b2          = (const float*)d_in[8];

    const int* src = edge_index;
    const int* dst = edge_index + N_EDGES;

    float* ws   = (float*)d_ws;
    float* dinv = ws;                               // 50000 f32 (deg -> dinv)
    float* X    = ws + 50048;                       // [N][128] feature buffer
    float* T    = X + (size_t)N_NODES * D;          // [N][128] GEMM temp
    float* out  = (float*)d_out;

    const int ELEMS = N_NODES * D;                  // 6,400,000
    const size_t emb_lds = (size_t)D * EMB_S * sizeof(float);   // 133,120 B
    const size_t lay_lds = (size_t)D * LAY_S * sizeof(float);   //  67,584 B

    // 1) degree -> dinv
    zero_f_kernel<<<(N_NODES + 255) / 256, 256, 0, stream>>>(dinv, N_NODES);
    degree_kernel<<<(N_EDGES + 255) / 256, 256, 0, stream>>>(dst, dinv);
    dinv_kernel<<<(N_NODES + 255) / 256, 256, 0, stream>>>(dinv);

    // 2) X = embed_table[tokens] @ W_node^T + b_node
    embed_gemm_kernel<<<GEMM_GRID, 256, emb_lds, stream>>>(node_tokens, embed_table,
                                                           W_node_w, W_node_b, X);

    // 3) layer 1: T = X @ w1^T ; X = dinv^2*T + b1 ; edge scatter into X
    node_gemm_kernel<false><<<GEMM_GRID, 256, lay_lds, stream>>>(X, w1, T);
    agg_init_kernel<<<ELEMS / 256, 256, 0, stream>>>(T, dinv, b1, X);
    edge_scatter_kernel<<<(N_EDGES * 32) / 256, 256, 0, stream>>>(src, dst, dinv, T, X);

    // 4) layer 2: T = relu(X) @ w2^T ; out = dinv^2*T + b2 ; edge scatter into out
    node_gemm_kernel<true><<<GEMM_GRID, 256, lay_lds, stream>>>(X, w2, T);
    agg_init_kernel<<<ELEMS / 256, 256, 0, stream>>>(T, dinv, b2, out);
    edge_scatter_kernel<<<(N_EDGES * 32) / 256, 256, 0, stream>>>(src, dst, dinv, T, out);
}